// NMR_80556406604675
// MI455X (gfx1250) — compile-verified
//
#include <hip/hip_runtime.h>
#include <hip/hip_bf16.h>
#include <cstddef>

// Soft-silhouette renderer for MI455X (gfx1250).
// Edge distances d = kb*px + ka*py + kc are evaluated as a K=3 GEMM on the
// matrix pipe via V_WMMA_F32_16X16X4_F32 (16 edge-rows x 16 pixel-cols per op);
// the log-sigmoid / log1p chain runs on the VALU (transcendental pipe).

typedef float v2f __attribute__((ext_vector_type(2)));
typedef float v8f __attribute__((ext_vector_type(8)));

#define IMG_S      320
#define N_FACES    1538
#define N_TILES    385            // ceil(1538/4) faces, 4 faces per WMMA tile
#define FP_FACES   (N_TILES * 4)  // 1540 padded
#define N_VERTS    778
#define N_BATCH    4
#define INV_SIGMA  100.0f
#define N_STRIPS   ((IMG_S * IMG_S) / 16)   // 6400 16-pixel strips per batch

// ---------------------------------------------------------------------------
// Phase 1: per-face edge coefficients, stored WMMA-A-ready.
// Tile layout (64 floats): row r (0..15) holds [kb, ka, kc, 0].
// Rows 0-2: face q=0, 3-5: q=1, 8-10: q=2, 11-13: q=3; rows 6,7,14,15 unused.
// ---------------------------------------------------------------------------
__global__ void silh_coeffs(const float* __restrict__ verts,
                            const int*   __restrict__ faces,
                            float*       __restrict__ cws) {
  int idx = blockIdx.x * blockDim.x + threadIdx.x;
  if (idx >= N_BATCH * FP_FACES) return;
  int b = idx / FP_FACES;
  int f = idx % FP_FACES;
  int tile = f >> 2;
  int q    = f & 3;
  int row  = (q < 2) ? q * 3 : 8 + (q - 2) * 3;
  float* tbase = cws + (size_t)(b * N_TILES + tile) * 64;
  float* dst   = tbase + row * 4;

  // Fill the unused rows (never read from D, but keep them finite).
  if (q == 1 || q == 3) {
    float* pad = tbase + ((q == 1) ? 6 : 14) * 4;
    for (int r = 0; r < 2; ++r) {
      pad[r * 4 + 0] = 0.0f; pad[r * 4 + 1] = 0.0f;
      pad[r * 4 + 2] = -1e18f; pad[r * 4 + 3] = 0.0f;
    }
  }

  if (f >= N_FACES) {           // padded face -> prob == 0 contribution
    for (int r = 0; r < 3; ++r) {
      dst[r * 4 + 0] = 0.0f; dst[r * 4 + 1] = 0.0f;
      dst[r * 4 + 2] = -1e18f; dst[r * 4 + 3] = 0.0f;
    }
    return;
  }

  int i0 = faces[f * 3 + 0], i1 = faces[f * 3 + 1], i2 = faces[f * 3 + 2];
  const float* vb = verts + (size_t)b * N_VERTS * 3;
  float x0 = vb[i0 * 3 + 0], y0 = -vb[i0 * 3 + 1];   // y is negated
  float x1 = vb[i1 * 3 + 0], y1 = -vb[i1 * 3 + 1];
  float x2 = vb[i2 * 3 + 0], y2 = -vb[i2 * 3 + 1];

  float area2 = (x1 - x0) * (y2 - y0) - (y1 - y0) * (x2 - x0);
  float sgn = (area2 > 0.0f) ? 1.0f : ((area2 < 0.0f) ? -1.0f : 0.0f);

  float ax[3] = {x0, x1, x2}, ay[3] = {y0, y1, y2};
  float bx[3] = {x1, x2, x0}, by[3] = {y1, y2, y0};
  for (int e = 0; e < 3; ++e) {
    float ex = bx[e] - ax[e];
    float ey = by[e] - ay[e];
    float inv = 1.0f / (sqrtf(ex * ex + ey * ey) + 1e-8f);
    float s = sgn * inv * INV_SIGMA;
    dst[e * 4 + 0] = -s * ey;                       // px coefficient (K=0)
    dst[e * 4 + 1] =  s * ex;                       // py coefficient (K=1)
    dst[e * 4 + 2] =  s * (ey * ax[e] - ex * ay[e]);// constant       (K=2)
    dst[e * 4 + 3] = 0.0f;                          // K=3 unused
  }
}

// ---------------------------------------------------------------------------
// Phase 2: one wave per (batch, 16-pixel strip); loop over 385 face tiles.
// ---------------------------------------------------------------------------
__device__ __forceinline__ float lsig(float x) {
  // stable log_sigmoid(x) = min(x,0) - log1p(exp(-|x|))
  return fminf(x, 0.0f) - __logf(1.0f + __expf(-fabsf(x)));
}

__device__ __forceinline__ float log1m_prob(float lp) {
  float p = fminf(__expf(lp), 1.0f - 1e-6f);   // clip(exp(logp), 0, 1-1e-6)
  return __logf(1.0f - p);                      // log1p(-prob)
}

__global__ void __launch_bounds__(256)
silh_render(const float* __restrict__ cws, float* __restrict__ out) {
  int lane = threadIdx.x & 31;
  int wave = (blockIdx.x * blockDim.x + threadIdx.x) >> 5;
  if (wave >= N_BATCH * N_STRIPS) return;       // exact grid: never taken
  int b     = wave / N_STRIPS;
  int strip = wave % N_STRIPS;

  int n    = lane & 15;                         // pixel within strip
  int pidx = strip * 16 + n;                    // S==320: strips never split rows
  float fx = ((float)(pidx % IMG_S) + 0.5f) * (2.0f / IMG_S) - 1.0f;
  float fy = ((float)(pidx / IMG_S) + 0.5f) * (2.0f / IMG_S) - 1.0f;

  // B (4x16): col n = [px, py, 1, 0]^T.
  // Lanes 0-15 hold {K0,K1}={px,py}; lanes 16-31 hold {K2,K3}={1,0}.
  v2f bm;
  bm.x = (lane < 16) ? fx : 1.0f;
  bm.y = (lane < 16) ? fy : 0.0f;

  // A: lanes 0-15 load {kb,ka} of row=lane; lanes 16-31 load {kc,0} of row=lane-16.
  const float* At = cws + (size_t)b * N_TILES * 64
                        + (size_t)((lane & 15) * 4 + (lane >> 4) * 2);

  float acc = 0.0f;
  for (int t = 0; t < N_TILES; ++t) {
    v2f a = *(const v2f*)(At + (size_t)t * 64);
    if (t + 8 < N_TILES) __builtin_prefetch(At + (size_t)(t + 8) * 64, 0, 1);

    v8f d = {};
    // D[edge_row, pixel_col] = kb*px + ka*py + kc  (scaled by 1/sigma already)
    d = __builtin_amdgcn_wmma_f32_16x16x4_f32(
            /*neg_a=*/false, a, /*neg_b=*/false, bm,
            /*c_mod=*/(short)0, d, /*reuse_a=*/false, /*reuse_b=*/false);

    // lanes 0-15: d[0..5] = rows 0-5 (faces 0,1); lanes 16-31: rows 8-13 (faces 2,3)
    float lp0 = lsig(d[0]) + lsig(d[1]) + lsig(d[2]);
    float lp1 = lsig(d[3]) + lsig(d[4]) + lsig(d[5]);
    acc += log1m_prob(lp0) + log1m_prob(lp1);
  }

  // combine the two face-halves of each pixel (lane n <-> lane n+16)
  float tot = acc + __shfl_xor(acc, 16, 32);
  if (lane < 16) out[(size_t)b * (IMG_S * IMG_S) + pidx] = 1.0f - __expf(tot);
}

// ---------------------------------------------------------------------------
extern "C" void kernel_launch(void* const* d_in, const int* in_sizes, int n_in,
                              void* d_out, int out_size, void* d_ws, size_t ws_size,
                              hipStream_t stream) {
  const float* verts = (const float*)d_in[0];   // (4,778,3) f32
  const int*   faces = (const int*)d_in[1];     // (1538,3) i32
  float*       out   = (float*)d_out;           // (4,320,320) f32
  float*       cws   = (float*)d_ws;            // 4*385*64 floats = 394240 B

  {
    int total = N_BATCH * FP_FACES;
    silh_coeffs<<<(total + 255) / 256, 256, 0, stream>>>(verts, faces, cws);
  }
  {
    int total_threads = N_BATCH * N_STRIPS * 32;   // 819200
    silh_render<<<total_threads / 256, 256, 0, stream>>>(cws, out);
  }
}